// PositionalAttention_51402168599369
// MI455X (gfx1250) — compile-verified
//
#include <hip/hip_runtime.h>
#include <hip/hip_bf16.h>
#include <math.h>

typedef float v2f __attribute__((ext_vector_type(2)));
typedef float v8f __attribute__((ext_vector_type(8)));

#define Bb   64
#define Ss   1024
#define Dd   128
#define Hh   20
#define GN   80     // 4*H

// ---------------------------------------------------------------------------
// Kernel 1: gates_pre[b*S+s][g] = sum_d x[b,s,d]*W_ih[g,d] + (b_ih[g]+b_hh[g])
// WMMA f32 16x16x4.  Grid: 65536 rows / 64 rows-per-block = 1024 blocks, 128 thr.
// ---------------------------------------------------------------------------
__global__ void k_gates(const float* __restrict__ x,
                        const float* __restrict__ Wih,
                        const float* __restrict__ bih,
                        const float* __restrict__ bhh,
                        float* __restrict__ gp) {
    __shared__ float Wt[Dd][GN];                 // W_ih transposed: Wt[d][g], 40 KB
    const int tid  = threadIdx.x;
    const int wave = tid >> 5;
    const int lane = tid & 31;

    for (int e = tid; e < GN * Dd; e += 128) {   // coalesced read, transpose into LDS
        int g = e >> 7, d = e & 127;
        Wt[d][g] = Wih[e];
    }
    __syncthreads();

    const int rowbase = blockIdx.x * 64 + wave * 16;
    const int M  = lane & 15;
    const int kh = (lane >> 4) * 2;              // 0 or 2
    const float* xrow = x + (size_t)(rowbase + M) * Dd;

    v8f acc[5];
    #pragma unroll
    for (int nb = 0; nb < 5; ++nb) acc[nb] = (v8f){0,0,0,0,0,0,0,0};

    for (int kk = 0; kk < Dd / 4; ++kk) {
        const int k0 = 4 * kk + kh;
        v2f a; a.x = xrow[k0]; a.y = xrow[k0 + 1];
        #pragma unroll
        for (int nb = 0; nb < 5; ++nb) {
            v2f bf; bf.x = Wt[k0][nb * 16 + M]; bf.y = Wt[k0 + 1][nb * 16 + M];
            acc[nb] = __builtin_amdgcn_wmma_f32_16x16x4_f32(
                false, a, false, bf, (short)0, acc[nb], false, false);
        }
    }

    #pragma unroll
    for (int nb = 0; nb < 5; ++nb) {
        const int g = nb * 16 + M;
        const float bias = bih[g] + bhh[g];
        #pragma unroll
        for (int r = 0; r < 8; ++r) {
            const int row = rowbase + (lane >> 4) * 8 + r;
            gp[(size_t)row * GN + g] = acc[nb][r] + bias;
        }
    }
}

// ---------------------------------------------------------------------------
// Kernel 2: sequential LSTM.  One block per batch, 128 threads (80 gate lanes).
// ---------------------------------------------------------------------------
__global__ void k_lstm(const float* __restrict__ gp,
                       const float* __restrict__ Whh,
                       float* __restrict__ hs) {
    __shared__ float sW[GN * Hh];
    __shared__ float h[Hh], c[Hh];
    __shared__ float sg[GN];
    const int b = blockIdx.x, tid = threadIdx.x;

    for (int e = tid; e < GN * Hh; e += 128) sW[e] = Whh[e];
    if (tid < Hh) { h[tid] = 0.f; c[tid] = 0.f; }
    __syncthreads();

    const float* gpb = gp + (size_t)b * Ss * GN;
    float* hsb = hs + (size_t)b * Ss * Hh;

    for (int t = 0; t < Ss; ++t) {
        if (tid < GN) {
            float v = gpb[(size_t)t * GN + tid];
            const float* wr = sW + tid * Hh;
            #pragma unroll
            for (int k = 0; k < Hh; ++k) v += wr[k] * h[k];
            sg[tid] = v;
        }
        __syncthreads();
        if (tid < Hh) {
            const float si = 1.f / (1.f + __expf(-sg[tid]));
            const float sf = 1.f / (1.f + __expf(-sg[20 + tid]));
            const float so = 1.f / (1.f + __expf(-sg[60 + tid]));
            const float cn = sf * c[tid] + si * tanhf(sg[40 + tid]);
            const float hn = so * tanhf(cn);
            c[tid] = cn; h[tid] = hn;
            hsb[(size_t)t * Hh + tid] = hn;
        }
        __syncthreads();
    }
}

// ---------------------------------------------------------------------------
// Kernel 3: mu_w / sigma heads + sequential mu scan.  One block per batch.
// ---------------------------------------------------------------------------
__global__ void k_head(const float* __restrict__ hs,
                       const float* __restrict__ Wmu,  const float* __restrict__ bmu,
                       const float* __restrict__ Wsig, const float* __restrict__ bsig,
                       float* __restrict__ mu, float* __restrict__ sigma) {
    __shared__ float sa[Ss], sc[Ss];
    const int b = blockIdx.x, tid = threadIdx.x;     // 256 threads
    const float* hsb = hs + (size_t)b * Ss * Hh;

    for (int j = tid; j < Ss; j += 256) {
        const float* hv = hsb + (size_t)j * Hh;
        float m0 = bmu[0], m1 = bmu[1], m2 = bmu[2], sv = bsig[0];
        #pragma unroll
        for (int k = 0; k < Hh; ++k) {
            const float hk = hv[k];
            m0 += Wmu[k] * hk;
            m1 += Wmu[Hh + k] * hk;
            m2 += Wmu[2 * Hh + k] * hk;
            sv += Wsig[k] * hk;
        }
        m0 = fmaxf(m0, 0.f); m1 = fmaxf(m1, 0.f); m2 = fmaxf(m2, 0.f);
        sigma[(size_t)b * Ss + j] = 1.f / (1.f + __expf(-sv));
        sa[j] = m0;
        sc[j] = m1 * (1.f / (float)Ss) + m2 * (((float)j + 1.f) / (float)Ss);
    }
    __syncthreads();
    if (tid == 0) {                                   // sequential scan (cheap)
        float prev = 0.f;
        for (int j = 0; j < Ss; ++j) {
            prev = sa[j] * prev + sc[j];
            mu[(size_t)b * Ss + j] = prev;
        }
    }
}

// ---------------------------------------------------------------------------
// Kernel 4: out[b,j,:] = (w[j,:]/||w[j,:]||) @ x[b,:,:] with on-the-fly Gaussian w.
// Grid (S/16, B); 128 threads = 4 waves, each wave owns 2 d-tiles of 16.
// ---------------------------------------------------------------------------
__global__ void k_attn(const float* __restrict__ x,
                       const float* __restrict__ mu,
                       const float* __restrict__ sigma,
                       float* __restrict__ out) {
    __shared__ float sWt[16][16];       // w tile (j x t)
    __shared__ float sX[16][Dd];        // x tile (t x d), 8 KB
    __shared__ float sPart[16][8];
    __shared__ float sInv[16];

    const int tid  = threadIdx.x;
    const int lane = tid & 31;
    const int wave = tid >> 5;
    const int b    = blockIdx.y;
    const int jb   = blockIdx.x;
    const int jbase = jb * 16;

    const float* xb = x + (size_t)b * Ss * Dd;

    // w-generation mapping: row = tid/8, this thread covers 2 t-columns
    const int rowW = tid >> 3;
    const int pcol = (tid & 7) * 2;
    const int jrow = jbase + rowW;
    const float mj = mu[(size_t)b * Ss + jrow];
    const float sj = sigma[(size_t)b * Ss + jrow];
    const float inv2s2 = 1.f / (2.f * sj * sj);
    const float jr = 1.f / ((float)jrow + 1.f);
    float sumsq = 0.f;

    // WMMA fragment mapping
    const int M  = lane & 15;
    const int kh = (lane >> 4) * 2;
    const int dt0 = wave * 2, dt1 = wave * 2 + 1;

    v8f acc0 = (v8f){0,0,0,0,0,0,0,0};
    v8f acc1 = (v8f){0,0,0,0,0,0,0,0};

    const int nchunks = jb + 1;                       // causal: t <= jbase+15
    for (int ch = 0; ch < nchunks; ++ch) {
        const int tc = ch * 16;
        __syncthreads();                              // LDS reuse fence

        // stage x[b, tc:tc+16, :] as float4 (512 vec4 / 128 threads)
        const float4* src = (const float4*)(xb + (size_t)tc * Dd);
        float4* dst = (float4*)(&sX[0][0]);
        #pragma unroll
        for (int e = tid; e < 512; e += 128) dst[e] = src[e];

        // Gaussian weights for this (j,t) tile + masked sum of squares
        #pragma unroll
        for (int q = 0; q < 2; ++q) {
            const int t = tc + pcol + q;
            float val = 0.f;
            if (t <= jrow) {
                const float dlt = (float)t * jr - mj;
                val = __expf(-dlt * dlt * inv2s2);
            }
            sWt[rowW][pcol + q] = val;
            sumsq += val * val;
        }
        __syncthreads();

        #pragma unroll
        for (int kk = 0; kk < 4; ++kk) {
            const int k0 = 4 * kk + kh;
            v2f a;  a.x  = sWt[M][k0];            a.y  = sWt[M][k0 + 1];
            v2f b0; b0.x = sX[k0][dt0 * 16 + M];  b0.y = sX[k0 + 1][dt0 * 16 + M];
            acc0 = __builtin_amdgcn_wmma_f32_16x16x4_f32(
                false, a, false, b0, (short)0, acc0, false, false);
            v2f b1; b1.x = sX[k0][dt1 * 16 + M];  b1.y = sX[k0 + 1][dt1 * 16 + M];
            acc1 = __builtin_amdgcn_wmma_f32_16x16x4_f32(
                false, a, false, b1, (short)0, acc1, false, false);
        }
    }

    // deterministic tree reduction of per-row sum of squares
    sPart[rowW][tid & 7] = sumsq;
    __syncthreads();
    if (tid < 16) {
        float s = 0.f;
        #pragma unroll
        for (int p = 0; p < 8; ++p) s += sPart[tid][p];
        sInv[tid] = 1.f / fmaxf(sqrtf(s), 1e-12f);
    }
    __syncthreads();

    float* ob = out + (size_t)b * Ss * Dd + (size_t)jbase * Dd;
    #pragma unroll
    for (int r = 0; r < 8; ++r) {
        const int Mr = (lane >> 4) * 8 + r;
        const float inv = sInv[Mr];
        ob[(size_t)Mr * Dd + dt0 * 16 + M] = acc0[r] * inv;
        ob[(size_t)Mr * Dd + dt1 * 16 + M] = acc1[r] * inv;
    }
}

// ---------------------------------------------------------------------------
extern "C" void kernel_launch(void* const* d_in, const int* in_sizes, int n_in,
                              void* d_out, int out_size, void* d_ws, size_t ws_size,
                              hipStream_t stream) {
    const float* x     = (const float*)d_in[0];
    const float* Wih   = (const float*)d_in[1];
    const float* Whh   = (const float*)d_in[2];
    const float* bih   = (const float*)d_in[3];
    const float* bhh   = (const float*)d_in[4];
    const float* Wmu   = (const float*)d_in[5];
    const float* bmu   = (const float*)d_in[6];
    const float* Wsig  = (const float*)d_in[7];
    const float* bsig  = (const float*)d_in[8];
    float* out = (float*)d_out;

    float* ws    = (float*)d_ws;
    float* gp    = ws;                                  // 65536*80
    float* hs    = gp + (size_t)Bb * Ss * GN;           // 65536*20
    float* muv   = hs + (size_t)Bb * Ss * Hh;           // 65536
    float* sig   = muv + (size_t)Bb * Ss;               // 65536

    k_gates<<<dim3((Bb * Ss) / 64), dim3(128), 0, stream>>>(x, Wih, bih, bhh, gp);
    k_lstm <<<dim3(Bb),            dim3(128), 0, stream>>>(gp, Whh, hs);
    k_head <<<dim3(Bb),            dim3(256), 0, stream>>>(hs, Wmu, bmu, Wsig, bsig, muv, sig);
    k_attn <<<dim3(Ss / 16, Bb),   dim3(128), 0, stream>>>(x, muv, sig, out);
}